// SeeSawLoss_45956150067469
// MI455X (gfx1250) — compile-verified
//
#include <hip/hip_runtime.h>

// Problem geometry (fixed by the reference): B=8, N=128, H=W=128
#define BB 8
#define NN 128
#define HWD 16384                          // H*W
#define THREADS 256                        // 8 waves of 32
#define PIXT 2                             // pixels per thread (float2 / b64 loads)
#define PIX_PER_BLOCK (THREADS * PIXT)     // 512
#define BLOCKS_PER_B (HWD / PIX_PER_BLOCK) // 32
#define NBLOCKS (BB * BLOCKS_PER_B)        // 256
#define PFDIST 16                          // prefetch distance in i-iterations (1 MiB ahead)

typedef __attribute__((ext_vector_type(2))) float v2f;
typedef __attribute__((ext_vector_type(8))) float v8f;

// Full-precision wave32 sum using V_WMMA_F32_16X16X4_F32.
// A (16x4): lane m holds {k0=v, k1=0}; lane m+16 holds {k2=v', k3=0}.
// B = all ones (4x16)  =>  D[m,n] = p[m] + p[m+16] for every n.
// D layout: VGPR r, lanes 0-15 -> M=r ; lanes 16-31 -> M=8+r.
// Summing the 8 D registers gives half-sums per half-wave; one xor-16 shuffle
// completes the full 32-lane sum (all lanes hold the total). EXEC must be all 1s.
__device__ __forceinline__ float wave_sum32_wmma(float v) {
    v2f a; a[0] = v;    a[1] = 0.0f;
    v2f b; b[0] = 1.0f; b[1] = 1.0f;
    v8f c = {};
    v8f d = __builtin_amdgcn_wmma_f32_16x16x4_f32(
        /*neg_a=*/false, a, /*neg_b=*/false, b,
        /*c_mod=*/(short)0, c, /*reuse_a=*/false, /*reuse_b=*/false);
    float s = ((d[0] + d[1]) + (d[2] + d[3])) + ((d[4] + d[5]) + (d[6] + d[7]));
    s += __shfl_xor(s, 16, 32);
    return s;
}

__global__ __launch_bounds__(THREADS) void seesaw_main_kernel(
    const float* __restrict__ logit,       // [B, N, H, W] f32
    const long long* __restrict__ target,  // [B, H, W] i64
    const float* __restrict__ weight,      // [B, N] f32
    float* __restrict__ partials)          // [NBLOCKS] f32 (in d_ws)
{
    __shared__ float sw[NN];
    __shared__ float wavesum[THREADS / 32];

    const int b   = blockIdx.x / BLOCKS_PER_B;
    const int blk = blockIdx.x % BLOCKS_PER_B;

    if (threadIdx.x < NN) sw[threadIdx.x] = weight[b * NN + threadIdx.x];
    __syncthreads();

    const int pix0 = (blk * THREADS + (int)threadIdx.x) * PIXT;

    // Per-pixel target class and seesaw scale base
    const long long* tp = target + (size_t)b * HWD + pix0;
    const int t0 = (int)tp[0];
    const int t1 = (int)tp[1];
    const float wt0 = sw[t0], wt1 = sw[t1];
    const float iwt0 = 1.0f / wt0, iwt1 = 1.0f / wt1;

    const float* base = logit + (size_t)b * NN * HWD;
    const v2f*   lp   = (const v2f*)base + (pix0 >> 1);   // stride HWD/2 v2f per class

    float denom0 = 0.0f, denom1 = 0.0f;
#pragma unroll 4
    for (int i = 0; i < NN; ++i) {
        v2f x = lp[(size_t)i * (HWD / 2)];                // coalesced b64 stream
        if (i + PFDIST < NN)                              // gfx1250 global_prefetch_b8
            __builtin_prefetch((const void*)(lp + (size_t)(i + PFDIST) * (HWD / 2)), 0, 3);
        const float wi = sw[i];
        const float e0 = __expf(x[0]);
        const float e1 = __expf(x[1]);
        const float f0 = (wi > wt0) ? wi * iwt0 : 1.0f;   // M[i, t] on the fly
        const float f1 = (wi > wt1) ? wi * iwt1 : 1.0f;
        denom0 = fmaf(e0, f0, denom0);
        denom1 = fmaf(e1, f1, denom1);
    }

    // -log(num/denom) = log(denom) - logit_t ; logit_t gather hits L2 (array resident)
    const float lt0 = base[(size_t)t0 * HWD + pix0];
    const float lt1 = base[(size_t)t1 * HWD + pix0 + 1];
    float loss = (__logf(denom0) - lt0) + (__logf(denom1) - lt1);

    // Wave-level reduction via WMMA (all 256 threads active -> EXEC all ones)
    const float wsum = wave_sum32_wmma(loss);
    const int lane = threadIdx.x & 31;
    const int wid  = threadIdx.x >> 5;
    if (lane == 0) wavesum[wid] = wsum;
    __syncthreads();

    if (threadIdx.x == 0) {
        float s = 0.0f;
#pragma unroll
        for (int w = 0; w < THREADS / 32; ++w) s += wavesum[w];
        partials[blockIdx.x] = s;   // fixed order -> deterministic
    }
}

__global__ __launch_bounds__(32) void seesaw_finalize_kernel(
    const float* __restrict__ partials, float* __restrict__ out)
{
    // Single wave, all 32 lanes active; fixed-order strided sum -> deterministic
    float s = 0.0f;
    for (int j = (int)threadIdx.x; j < NBLOCKS; j += 32) s += partials[j];
    s = wave_sum32_wmma(s);
    if (threadIdx.x == 0) out[0] = s * (1.0f / ((float)BB * (float)HWD));
}

extern "C" void kernel_launch(void* const* d_in, const int* in_sizes, int n_in,
                              void* d_out, int out_size, void* d_ws, size_t ws_size,
                              hipStream_t stream) {
    (void)in_sizes; (void)n_in; (void)out_size; (void)ws_size;
    const float*     logit  = (const float*)d_in[0];
    const long long* target = (const long long*)d_in[1];
    const float*     weight = (const float*)d_in[2];
    // d_in[3] = epoch (unused by the math)
    float* partials = (float*)d_ws;   // 256 floats of scratch

    seesaw_main_kernel<<<NBLOCKS, THREADS, 0, stream>>>(logit, target, weight, partials);
    seesaw_finalize_kernel<<<1, 32, 0, stream>>>(partials, (float*)d_out);
}